// ShiftInvariantLoss_77077483094885
// MI455X (gfx1250) — compile-verified
//
#include <hip/hip_runtime.h>
#include <hip/hip_bf16.h>
#include <math.h>

// ShiftInvariantLoss on MI455X (gfx1250), wave32 + V_WMMA_F32_16X16X4_F32.
//
// loss[b,s] = mean(max(l,0)+log1p(exp(-|l|)))  -  (1/16384) * corr[b,s]
// corr[b,i,j] = sum_{y,x} L[b,y,x] * T[b,y+i,x+j]
//
// Per logit row y the 9x9 shift grid is one 16x16 WMMA tile:
//   D[i,j] += sum_c A[i,c]*B[c,j],  A[i,c]=T[y+i,c] (K=136),
//   B[c,j]=L[y,c-j] (Toeplitz), accumulated over y in the WMMA accumulator.
// Zero-padding in LDS (cols for L, rows for T) removes all bounds checks so
// the inner loop is pure ds_load + v_wmma in uniform control flow.

typedef __attribute__((ext_vector_type(2))) float v2f;
typedef __attribute__((ext_vector_type(8))) float v8f;

constexpr int kB    = 64;            // batches
constexpr int kN    = 128;           // logit side
constexpr int kM    = 136;           // target side
constexpr int kSh   = 9;             // shifts per dim
constexpr int kNPix = kN * kN;       // 16384
constexpr int kTPix = kM * kM;       // 18496
constexpr int kKCh  = kM / 4;        // 34 K-steps of 4
constexpr int kCenter = (kSh / 2) * kSh + (kSh / 2);  // 40

// Padded logits: row stride 160, data at cols [16,144), zeros elsewhere.
// B index 16 + c0 + ks - r16 is in [1,151] -> always in-bounds.
constexpr int kLStride = 160;
constexpr int kLPad    = 16;
constexpr int kLpFloats = kN * kLStride;          // 20480
// Padded targets: 7 extra zero rows so row y + r16 <= 142 is in-bounds.
constexpr int kTRows   = kM + 7;                  // 143
constexpr int kTpFloats = kTRows * kM;            // 19448

// LDS (floats): sLp | sTp | sPart[8*256] | sCorr[256] | sRed[256]
constexpr int kSmemFloats = kLpFloats + kTpFloats + 8 * 256 + 256 + 256;

__global__ void __launch_bounds__(256) sil_corr_kernel(
    const float* __restrict__ L, const float* __restrict__ T,
    float* __restrict__ ws_loss, int* __restrict__ ws_idx)
{
  extern __shared__ float smem[];
  float* sLp   = smem;                  // padded logits
  float* sTp   = sLp + kLpFloats;       // padded targets
  float* sPart = sTp + kTpFloats;       // per-wave 16x16 partial tiles
  float* sCorr = sPart + 8 * 256;       // summed 16x16 corr tile
  float* sRed  = sCorr + 256;           // K(b) tree reduction

  const int b   = blockIdx.x;
  const int tid = threadIdx.x;          // 0..255 (8 waves of 32)
  const float* Lb = L + b * kNPix;
  const float* Tb = T + b * kTPix;

  // Stage logits with zeroed column padding.
  for (int i = tid; i < kLpFloats; i += 256) {
    const int y = i / kLStride;
    const int c = i - y * kLStride;
    const int x = c - kLPad;
    sLp[i] = (x >= 0 && x < kN) ? Lb[y * kN + x] : 0.f;
  }
  // Stage targets with zeroed row padding.
  for (int i = tid; i < kTpFloats; i += 256)
    sTp[i] = (i < kTPix) ? Tb[i] : 0.f;
  __syncthreads();

  const int wave = tid >> 5;
  const int lane = tid & 31;
  const int r16  = lane & 15;           // A row i / B col j / D col j
  const int hi   = lane >> 4;           // 0: K=0,1   1: K=2,3
  const int ks   = hi * 2;

  v8f acc = {0.f, 0.f, 0.f, 0.f, 0.f, 0.f, 0.f, 0.f};

  for (int y = wave; y < kN; y += 8) {
    // A rows: T[y + r16][*], 8-byte aligned pair at (.. + c0 + ks)
    const float* arow = sTp + (y + r16) * kM + ks;
    // B: L[y][c0 + ks - r16 + {0,1}] via padded row (no bounds checks)
    const float* brow = sLp + y * kLStride + kLPad + ks - r16;
    #pragma unroll 2
    for (int kk = 0; kk < kKCh; ++kk) {
      const int c0 = kk * 4;
      const v2f a = *(const v2f*)(arow + c0);   // ds_load_b64
      v2f bv;
      bv.x = brow[c0];
      bv.y = brow[c0 + 1];
      acc = __builtin_amdgcn_wmma_f32_16x16x4_f32(
          false, a, false, bv, (short)0, acc, false, false);
    }
  }

  // D layout: vgpr r, lane -> (i = r + 8*hi, j = r16)
  #pragma unroll
  for (int r = 0; r < 8; ++r)
    sPart[wave * 256 + (r + 8 * hi) * 16 + r16] = acc[r];

  // shift-independent BCE term: sum(max(l,0) + log1p(exp(-|l|)))
  float ksum = 0.f;
  for (int i = tid; i < kNPix; i += 256) {
    const int y = i >> 7;
    const int x = i & (kN - 1);
    const float v = sLp[y * kLStride + kLPad + x];
    ksum += fmaxf(v, 0.f) + log1pf(expf(-fabsf(v)));
  }
  sRed[tid] = ksum;
  __syncthreads();

  // deterministic fixed-order sums (no float atomics -> replay-stable)
  {
    float c = 0.f;
    #pragma unroll
    for (int w = 0; w < 8; ++w) c += sPart[w * 256 + tid];
    sCorr[tid] = c;
  }
  for (int s = 128; s > 0; s >>= 1) {
    __syncthreads();
    if (tid < s) sRed[tid] += sRed[tid + s];
  }
  __syncthreads();

  if (tid == 0) {
    const float inv   = 1.f / (float)kNPix;
    const float kmean = sRed[0] * inv;
    float minv = __builtin_inff();
    int   mini = 0;
    for (int s = 0; s < kSh * kSh; ++s) {
      const int i = s / kSh, j = s % kSh;
      const float loss = kmean - sCorr[i * 16 + j] * inv;
      if (loss < minv) { minv = loss; mini = s; }   // first-min tiebreak
    }
    const float lc = kmean - sCorr[4 * 16 + 4] * inv;
    if (lc == minv) mini = kCenter;                 // prefer center on tie
    ws_loss[b] = minv;
    ws_idx[b]  = mini;
  }
}

__global__ void sil_finalize_kernel(const float* __restrict__ ws_loss,
                                    const int* __restrict__ ws_idx,
                                    float* __restrict__ out)
{
  const int t = threadIdx.x;
  if (t == 0) {
    float tot = 0.f;
    for (int b = 0; b < kB; ++b) tot += ws_loss[b];  // fixed order
    out[0] = tot;
  }
  if (t < kB) {
    const int s = ws_idx[t];
    out[1 + kB * kNPix + t]      = (float)(s / kSh);  // row shifts
    out[1 + kB * kNPix + kB + t] = (float)(s % kSh);  // col shifts
  }
}

__global__ void __launch_bounds__(256) sil_gather_kernel(
    const float* __restrict__ T, const int* __restrict__ ws_idx,
    float* __restrict__ out)
{
  const int b = blockIdx.x;
  const int s = ws_idx[b];
  const float* Tb = T + b * kTPix + (s / kSh) * kM + (s % kSh);
  float* ob = out + 1 + b * kNPix;
  for (int t = threadIdx.x; t < kNPix; t += 256)
    ob[t] = Tb[(t >> 7) * kM + (t & (kN - 1))];
}

extern "C" void kernel_launch(void* const* d_in, const int* in_sizes, int n_in,
                              void* d_out, int out_size, void* d_ws, size_t ws_size,
                              hipStream_t stream) {
  (void)in_sizes; (void)n_in; (void)out_size; (void)ws_size;
  const float* L = (const float*)d_in[0];   // logits  (64,1,128,128) f32
  const float* T = (const float*)d_in[1];   // targets (64,1,136,136) f32
  float* out     = (float*)d_out;           // [1 | 64*16384 | 64 | 64] f32
  float* ws_loss = (float*)d_ws;            // 64 floats
  int*   ws_idx  = (int*)(ws_loss + kB);    // 64 ints

  const size_t smem = (size_t)kSmemFloats * sizeof(float);  // ~166 KB (<320 KB/WGP)
  sil_corr_kernel<<<dim3(kB), dim3(256), smem, stream>>>(L, T, ws_loss, ws_idx);
  sil_finalize_kernel<<<dim3(1), dim3(64), 0, stream>>>(ws_loss, ws_idx, out);
  sil_gather_kernel<<<dim3(kB), dim3(256), 0, stream>>>(T, ws_idx, out);
}